// DConvformerEncoder_62311385531190
// MI455X (gfx1250) — compile-verified
//
#include <hip/hip_runtime.h>
#include <math.h>

// ---------------------------------------------------------------------------
// DConvformerEncoder for MI455X (gfx1250, wave32).
// fp32 end-to-end; matrix work on V_WMMA_F32_16X16X4_F32 (fp32 tensor path).
// morph (max-plus) stays VALU; conv stack = implicit GEMM via WMMA; MLP = WMMA.
// Weights staged to LDS with gfx1250 async global->LDS (ASYNCcnt) copies.
// Activations (<250MB) fit in 192MB L2 -> im2col A gathers read global directly.
// ---------------------------------------------------------------------------

#define BDIM 256          // 8 waves of 32
#define NB   256          // batch
#define CTOK 33
#define DIM  64
#define IMG  (CTOK * DIM) // 2112 pixels per image
#define EPSV 1e-5f

typedef float v2f __attribute__((ext_vector_type(2)));
typedef float v8f __attribute__((ext_vector_type(8)));
typedef int   v4i __attribute__((ext_vector_type(4)));

// Async-to-LDS builtins take typed pointers: global (AS1) + LDS (AS3).
#define GASV4(p) ((__attribute__((address_space(1))) v4i*)(p))
#define LASV4(p) ((__attribute__((address_space(3))) v4i*)(p))
#define GASI(p)  ((__attribute__((address_space(1))) int*)(p))
#define LASI(p)  ((__attribute__((address_space(3))) int*)(p))

__device__ __forceinline__ float gelu_exact(float x) {
  return 0.5f * x * (1.0f + erff(x * 0.70710678118654752f));
}

// Cooperative global->LDS copy of n floats (n % 4 == 0, 16B-aligned both sides).
// Uses gfx1250 async-to-LDS path when available (ASYNCcnt), else plain copy.
__device__ __forceinline__ void async_copy_f32(float* lds_dst, const float* gsrc,
                                               int nfloats) {
#if __has_builtin(__builtin_amdgcn_global_load_async_to_lds_b128)
  for (int i = threadIdx.x * 4; i < nfloats; i += BDIM * 4)
    __builtin_amdgcn_global_load_async_to_lds_b128(GASV4(gsrc + i), LASV4(lds_dst + i), 0, 0);
#elif __has_builtin(__builtin_amdgcn_global_load_async_to_lds_b32)
  for (int i = threadIdx.x; i < nfloats; i += BDIM)
    __builtin_amdgcn_global_load_async_to_lds_b32(GASI(gsrc + i), LASI(lds_dst + i), 0, 0);
#else
  for (int i = threadIdx.x; i < nfloats; i += BDIM) lds_dst[i] = gsrc[i];
#endif
}

__device__ __forceinline__ void async_copy_fence() {
#if __has_builtin(__builtin_amdgcn_s_wait_asynccnt)
  __builtin_amdgcn_s_wait_asynccnt(0);
#endif
  __syncthreads();
}

// One 16x16 fp32 output tile per wave32, D = A(16xK) * B(Kx16) + D.
// loadA(m,k) / loadB(k,n) are bounds-guarded gathers (return 0 when OOB).
// Fragment layouts per CDNA5 ISA 7.12.2 (32-bit A 16x4; 32-bit C/D 16x16).
// No unroll pragma: constant Kpad (MLP) fully unrolls, runtime Kpad (conv) rolls.
template <typename FA, typename FB>
__device__ __forceinline__ void gemm_tile_16x16(float (&acc)[8], int Kpad,
                                                FA loadA, FB loadB) {
  const int lane = threadIdx.x & 31;
  const int half = lane >> 4;   // K-half for A fragment
  const int l15  = lane & 15;   // M row (A) / N col (B,C,D)
#if __has_builtin(__builtin_amdgcn_wmma_f32_16x16x4_f32)
  v8f c;
#pragma unroll
  for (int i = 0; i < 8; ++i) c[i] = acc[i];
  for (int kb = 0; kb < Kpad; kb += 4) {
    v2f a, b;
    a[0] = loadA(l15, kb + 2 * half + 0);
    a[1] = loadA(l15, kb + 2 * half + 1);
    b[0] = loadB(kb + half, l15);
    b[1] = loadB(kb + 2 + half, l15);
    c = __builtin_amdgcn_wmma_f32_16x16x4_f32(false, a, false, b,
                                              (short)0, c, false, false);
  }
#pragma unroll
  for (int i = 0; i < 8; ++i) acc[i] = c[i];
#else
  for (int k = 0; k < Kpad; ++k) {
    float b = loadB(k, l15);
#pragma unroll
    for (int i = 0; i < 8; ++i) acc[i] += loadA(i + 8 * half, k) * b;
  }
#endif
}

// ---------------------------------------------------------------------------
// Kernel 1: morph stage (erosion/dilation + small convs + BN + GELU). VALU.
// grid (NB, 2), block 256. xstage layout: [2][NB][33*64]
// ---------------------------------------------------------------------------
__global__ void morph_stage_kernel(
    const float* __restrict__ htok, const float* __restrict__ ltok,
    const float* __restrict__ ero_sp, const float* __restrict__ dil_sp,
    const float* __restrict__ c1w_sp, const float* __restrict__ c1b_sp,
    const float* __restrict__ c2w_sp, const float* __restrict__ c2b_sp,
    const float* __restrict__ bng_sp, const float* __restrict__ bnb_sp,
    const float* __restrict__ ero_sc, const float* __restrict__ dil_sc,
    const float* __restrict__ c1w_sc, const float* __restrict__ c1b_sc,
    const float* __restrict__ c2w_sc, const float* __restrict__ c2b_sc,
    const float* __restrict__ bng_sc, const float* __restrict__ bnb_sc,
    float* __restrict__ xstage) {
  const int b = blockIdx.x, t = blockIdx.y, tid = threadIdx.x;
  const float* src = (t ? ltok : htok) + (size_t)b * IMG;
  float* dst = xstage + ((size_t)t * NB + b) * IMG;

  __shared__ float xin[32 * 64];     // channels 1..32 of the 8x8 tiles
  __shared__ float mo[4][16 * 64];   // ero_sp, dil_sp, ero_sc, dil_sc

  async_copy_f32(xin, src + 64, 2048);                   // async global->LDS
  for (int i = tid; i < 64; i += BDIM) dst[i] = src[i];  // channel 0 passthrough
  async_copy_fence();

  const float* mw[4] = {ero_sp, dil_sp, ero_sc, dil_sc};
  for (int m = 0; m < 4; ++m) {
    const bool ero = (m & 1) == 0;
    const float* W = mw[m];
    for (int item = tid; item < 1024; item += BDIM) {
      int oc = item >> 6, p = item & 63;
      int h8 = p >> 3, w8 = p & 7;
      float best = -3.4e38f;
      for (int c = 0; c < 32; ++c)
        for (int ki = 0; ki < 3; ++ki)
          for (int kj = 0; kj < 3; ++kj) {
            int ih = h8 + ki - 1, iw = w8 + kj - 1;
            float pv = (ih >= 0 && ih < 8 && iw >= 0 && iw < 8)
                           ? xin[c * 64 + ih * 8 + iw] : 0.0f;
            float wv = W[oc * 288 + c * 9 + ki * 3 + kj];
            float cand = ero ? (wv - pv) : (wv + pv);
            best = fmaxf(best, cand);
          }
      mo[m][item] = ero ? -best : best;
    }
  }
  __syncthreads();

  const float bnscale = rsqrtf(1.0f + EPSV);
  for (int item = tid; item < 1024; item += BDIM) {
    int oc = item >> 6, p = item & 63;
    int h8 = p >> 3, w8 = p & 7;
    // spat branch: 3x3 conv pad 1 on ero/dil results, sum, BN, GELU
    float z = c1b_sp[oc] + c2b_sp[oc];
    for (int ic = 0; ic < 16; ++ic)
      for (int ki = 0; ki < 3; ++ki)
        for (int kj = 0; kj < 3; ++kj) {
          int ih = h8 + ki - 1, iw = w8 + kj - 1;
          if (ih < 0 || ih >= 8 || iw < 0 || iw >= 8) continue;
          int wi = oc * 144 + ic * 9 + ki * 3 + kj;
          z += mo[0][ic * 64 + ih * 8 + iw] * c1w_sp[wi]
             + mo[1][ic * 64 + ih * 8 + iw] * c2w_sp[wi];
        }
    z = z * bng_sp[oc] * bnscale + bnb_sp[oc];
    dst[(1 + oc) * 64 + p] = gelu_exact(z);
    // spec branch: 1x1 convs
    float z2 = c1b_sc[oc] + c2b_sc[oc];
    for (int ic = 0; ic < 16; ++ic)
      z2 += mo[2][ic * 64 + p] * c1w_sc[oc * 16 + ic]
          + mo[3][ic * 64 + p] * c2w_sc[oc * 16 + ic];
    z2 = z2 * bng_sc[oc] * bnscale + bnb_sc[oc];
    dst[(17 + oc) * 64 + p] = gelu_exact(z2);
  }
}

// ---------------------------------------------------------------------------
// Kernel 2: LayerNorm over DIM=64 (one wave32 per row, 2 elems/lane). grid (NB,2)
// ---------------------------------------------------------------------------
__global__ void ln_kernel(const float* __restrict__ x, const float* __restrict__ g,
                          const float* __restrict__ bt, float* __restrict__ out) {
  const int b = blockIdx.x, t = blockIdx.y;
  const size_t base = ((size_t)t * NB + b) * IMG;
  const int wave = threadIdx.x >> 5, lane = threadIdx.x & 31;
  for (int row = wave; row < CTOK; row += 8) {
    float v0 = x[base + row * 64 + lane];
    float v1 = x[base + row * 64 + 32 + lane];
    float s = v0 + v1, q = v0 * v0 + v1 * v1;
    for (int off = 16; off > 0; off >>= 1) {
      s += __shfl_xor(s, off, 32);
      q += __shfl_xor(q, off, 32);
    }
    float mean = s * (1.0f / 64.0f);
    float var  = q * (1.0f / 64.0f) - mean * mean;
    float inv  = rsqrtf(var + EPSV);
    out[base + row * 64 + lane]      = (v0 - mean) * inv * g[lane] + bt[lane];
    out[base + row * 64 + 32 + lane] = (v1 - mean) * inv * g[lane + 32] + bt[lane + 32];
  }
}

// ---------------------------------------------------------------------------
// Kernel 3: dilated conv layer as implicit GEMM on WMMA fp32 tensor path.
// K is tap-major: k = tap*Cin + c, Cin = 1<<lc (all layers power-of-2), so the
// im2col decode is shift/mask (no integer division in the WMMA feed loop).
// Weights staged to LDS via async global->LDS; A gathered from L2-resident
// activations. One wave per 16(pixels) x 16(out-ch) tile. grid (gx, NB, 2).
// lnt = log2(#N tiles) (0 or 1) so the tile decode is shift/mask too.
// ---------------------------------------------------------------------------
__global__ void conv_wmma_kernel(const float* __restrict__ inbuf,
                                 float* __restrict__ outbuf,
                                 const float* __restrict__ w,
                                 const float* __restrict__ bias,
                                 const float* __restrict__ g,
                                 const float* __restrict__ bb,
                                 int lc, int Cout, int dil, int lnt) {
  const int b = blockIdx.y, t = blockIdx.z;
  const int Cin = 1 << lc;
  const int K = 9 << lc, Kpad = (K + 3) & ~3;

  __shared__ float wlds[32 * 144];   // up to 18.4KB of weights (Cout*K)
  async_copy_f32(wlds, w, Cout * K); // async global->LDS, all threads
  async_copy_fence();

  const int wave = threadIdx.x >> 5;
  const int tile = blockIdx.x * 8 + wave;
  if (tile >= (132 << lnt)) return;  // whole-wave uniform exit (after barrier)
  const int mtile = tile >> lnt;
  const int nb16 = (tile & ((1 << lnt) - 1)) * 16;
  const float* inp = inbuf + ((size_t)t * NB + b) * ((size_t)Cin * IMG);
  float* outp      = outbuf + ((size_t)t * NB + b) * ((size_t)Cout * IMG);

  __builtin_prefetch(inp + mtile * 16, 0, 3);  // global_prefetch of A plane

  auto loadA = [&](int m, int k) -> float {    // im2col gather with zero pad
    if (k >= K) return 0.0f;
    int tap = k >> lc, c = k & (Cin - 1);
    int ki = tap / 3, kj = tap - ki * 3;       // tap < 9: cheap const-div
    int p = mtile * 16 + m;
    int ih = (p >> 6) + (ki - 1) * dil, iw = (p & 63) + (kj - 1) * dil;
    if (ih < 0 || ih >= CTOK || iw < 0 || iw >= 64) return 0.0f;
    return inp[(size_t)c * IMG + ih * 64 + iw];
  };
  auto loadB = [&](int k, int n) -> float {    // weights from LDS
    int oc = nb16 + n;
    if (k >= K || oc >= Cout) return 0.0f;
    int tap = k >> lc, c = k & (Cin - 1);
    return wlds[(oc * Cin + c) * 9 + tap];
  };

  float acc[8] = {0, 0, 0, 0, 0, 0, 0, 0};
  gemm_tile_16x16(acc, Kpad, loadA, loadB);

  const int lane = threadIdx.x & 31;
  const int half = lane >> 4, l15 = lane & 15;
  const int oc = nb16 + l15;
  const float bnscale = rsqrtf(1.0f + EPSV);
  if (oc < Cout) {
    float bi = bias[oc], gg = g[oc] * bnscale, bbv = bb[oc];
#pragma unroll
    for (int i = 0; i < 8; ++i) {
      int p = mtile * 16 + i + 8 * half;
      float v = (acc[i] + bi) * gg + bbv;          // bias -> BN(eval)
      outp[(size_t)oc * IMG + p] = fmaxf(v, 0.0f); // ReLU
    }
  }
}

// ---------------------------------------------------------------------------
// Kernel 4: residual + token-0 swap + LN2 + MLP (two 64x64 WMMA GEMMs) + out.
// Both weight matrices async-staged to LDS. grid (NB, 2), block 256.
// ---------------------------------------------------------------------------
__global__ void mlp_kernel(const float* __restrict__ xstage,
                           const float* __restrict__ convout,
                           const float* __restrict__ g2, const float* __restrict__ b2v,
                           const float* __restrict__ w1, const float* __restrict__ bb1,
                           const float* __restrict__ w2, const float* __restrict__ bb2,
                           float* __restrict__ out) {
  const int b = blockIdx.x, t = blockIdx.y, tid = threadIdx.x;
  __shared__ float xr[IMG], hn[IMG], h1[IMG];
  __shared__ float w1l[DIM * DIM], w2l[DIM * DIM];

  async_copy_f32(w1l, w1, DIM * DIM);    // async global->LDS weight staging
  async_copy_f32(w2l, w2, DIM * DIM);
  // residual x + conv_module(ln(x)) with token-0 cross-stream swap
  for (int i = tid; i < IMG; i += BDIM) {
    int row = i >> 6;
    int tt = (row == 0) ? (1 - t) : t;
    size_t base = ((size_t)tt * NB + b) * IMG;
    xr[i] = xstage[base + i] + convout[base + i];
  }
  async_copy_fence();

  const int wave = tid >> 5, lane = tid & 31;
  for (int row = wave; row < CTOK; row += 8) {   // LN2
    float v0 = xr[row * 64 + lane], v1 = xr[row * 64 + 32 + lane];
    float s = v0 + v1, q = v0 * v0 + v1 * v1;
    for (int off = 16; off > 0; off >>= 1) {
      s += __shfl_xor(s, off, 32);
      q += __shfl_xor(q, off, 32);
    }
    float mean = s * (1.0f / 64.0f);
    float var  = q * (1.0f / 64.0f) - mean * mean;
    float inv  = rsqrtf(var + EPSV);
    hn[row * 64 + lane]      = (v0 - mean) * inv * g2[lane] + b2v[lane];
    hn[row * 64 + 32 + lane] = (v1 - mean) * inv * g2[lane + 32] + b2v[lane + 32];
  }
  __syncthreads();

  const int half = lane >> 4, l15 = lane & 15;
  // GEMM1 (33x64 @ 64x64) + bias + exact GELU -> h1
  for (int tile = wave; tile < 12; tile += 8) {
    int mt = tile >> 2, nb16 = (tile & 3) * 16;
    auto loadA = [&](int m, int k) -> float {
      int r = mt * 16 + m;
      return (r < CTOK) ? hn[r * 64 + k] : 0.0f;
    };
    auto loadB = [&](int k, int n) -> float { return w1l[k * 64 + nb16 + n]; };
    float acc[8] = {0, 0, 0, 0, 0, 0, 0, 0};
    gemm_tile_16x16(acc, 64, loadA, loadB);
    int n = nb16 + l15;
#pragma unroll
    for (int i = 0; i < 8; ++i) {
      int r = mt * 16 + i + 8 * half;
      if (r < CTOK) h1[r * 64 + n] = gelu_exact(acc[i] + bb1[n]);
    }
  }
  __syncthreads();
  // GEMM2 + bias + residual -> d_out
  size_t obase = ((size_t)t * NB + b) * IMG;
  for (int tile = wave; tile < 12; tile += 8) {
    int mt = tile >> 2, nb16 = (tile & 3) * 16;
    auto loadA = [&](int m, int k) -> float {
      int r = mt * 16 + m;
      return (r < CTOK) ? h1[r * 64 + k] : 0.0f;
    };
    auto loadB = [&](int k, int n) -> float { return w2l[k * 64 + nb16 + n]; };
    float acc[8] = {0, 0, 0, 0, 0, 0, 0, 0};
    gemm_tile_16x16(acc, 64, loadA, loadB);
    int n = nb16 + l15;
#pragma unroll
    for (int i = 0; i < 8; ++i) {
      int r = mt * 16 + i + 8 * half;
      if (r < CTOK) out[obase + r * 64 + n] = xr[r * 64 + n] + acc[i] + bb2[n];
    }
  }
}

// ---------------------------------------------------------------------------
// Host-side launch. Input order = jax pytree (dict keys sorted) flatten:
//  0:h_tokens 1:l_tokens
//  2..25: convmod[i] x {b, bb, g, w} for i=0..5
//  26:ln1_b 27:ln1_g 28:ln2_b 29:ln2_g 30:mlp_b1 31:mlp_b2 32:mlp_w1 33:mlp_w2
//  34..41: spat {bn_b,bn_g,c1b,c1w,c2b,c2w,dil_w,ero_w}
//  42..49: spec {bn_b,bn_g,c1b,c1w,c2b,c2w,dil_w,ero_w}
// ---------------------------------------------------------------------------
extern "C" void kernel_launch(void* const* d_in, const int* in_sizes, int n_in,
                              void* d_out, int out_size, void* d_ws, size_t ws_size,
                              hipStream_t stream) {
  (void)in_sizes; (void)n_in; (void)out_size; (void)ws_size;
  const float* htok = (const float*)d_in[0];
  const float* ltok = (const float*)d_in[1];
  const float *cm_b[6], *cm_bb[6], *cm_g[6], *cm_w[6];
  for (int i = 0; i < 6; ++i) {
    cm_b[i]  = (const float*)d_in[2 + 4 * i];
    cm_bb[i] = (const float*)d_in[3 + 4 * i];
    cm_g[i]  = (const float*)d_in[4 + 4 * i];
    cm_w[i]  = (const float*)d_in[5 + 4 * i];
  }
  const float* ln1_b = (const float*)d_in[26];
  const float* ln1_g = (const float*)d_in[27];
  const float* ln2_b = (const float*)d_in[28];
  const float* ln2_g = (const float*)d_in[29];
  const float* mb1 = (const float*)d_in[30];
  const float* mb2 = (const float*)d_in[31];
  const float* mw1 = (const float*)d_in[32];
  const float* mw2 = (const float*)d_in[33];
  const float* sp_bnb = (const float*)d_in[34];
  const float* sp_bng = (const float*)d_in[35];
  const float* sp_c1b = (const float*)d_in[36];
  const float* sp_c1w = (const float*)d_in[37];
  const float* sp_c2b = (const float*)d_in[38];
  const float* sp_c2w = (const float*)d_in[39];
  const float* sp_dil = (const float*)d_in[40];
  const float* sp_ero = (const float*)d_in[41];
  const float* sc_bnb = (const float*)d_in[42];
  const float* sc_bng = (const float*)d_in[43];
  const float* sc_c1b = (const float*)d_in[44];
  const float* sc_c1w = (const float*)d_in[45];
  const float* sc_c2b = (const float*)d_in[46];
  const float* sc_c2w = (const float*)d_in[47];
  const float* sc_dil = (const float*)d_in[48];
  const float* sc_ero = (const float*)d_in[49];

  float* ws = (float*)d_ws;
  const size_t per = (size_t)NB * IMG;          // 540,672 floats
  float* xstage = ws;                           // [2][NB][IMG]
  float* lnbuf  = xstage + 2 * per;             // [2][NB][1][IMG]
  float* bufA   = lnbuf + 2 * per;              // up to 32 channels
  float* bufB   = bufA + 2 * per * 32;          // up to 16 channels

  dim3 blk(BDIM);
  morph_stage_kernel<<<dim3(NB, 2), blk, 0, stream>>>(
      htok, ltok,
      sp_ero, sp_dil, sp_c1w, sp_c1b, sp_c2w, sp_c2b, sp_bng, sp_bnb,
      sc_ero, sc_dil, sc_c1w, sc_c1b, sc_c2w, sc_c2b, sc_bng, sc_bnb,
      xstage);
  ln_kernel<<<dim3(NB, 2), blk, 0, stream>>>(xstage, ln1_g, ln1_b, lnbuf);

  const int lcin[6] = {0, 3, 4, 5, 4, 3};       // log2(Cin): 1,8,16,32,16,8
  const int cout[6] = {8, 16, 32, 16, 8, 1};
  const int dils[6] = {1, 2, 5, 1, 2, 5};
  float* bufs[7] = {lnbuf, bufA, bufB, bufA, bufB, bufA, bufB};
  for (int i = 0; i < 6; ++i) {
    int lnt = (cout[i] > 16) ? 1 : 0;           // log2(#N tiles)
    int gx = ((132 << lnt) + 7) >> 3;
    conv_wmma_kernel<<<dim3(gx, NB, 2), blk, 0, stream>>>(
        bufs[i], bufs[i + 1], cm_w[i], cm_b[i], cm_g[i], cm_bb[i],
        lcin[i], cout[i], dils[i], lnt);
  }
  mlp_kernel<<<dim3(NB, 2), blk, 0, stream>>>(
      xstage, bufB, ln2_g, ln2_b, mw1, mb1, mw2, mb2, (float*)d_out);
}